// GMMGate_36421322670729
// MI455X (gfx1250) — compile-verified
//
#include <hip/hip_runtime.h>
#include <math.h>

// ---------------- problem constants (from reference) ----------------
#define S_TOKENS 8192
#define D_MODEL  1024
#define N_EXPERT 64
#define CAPACITY 256              // max(ceil(8192/64*2), 4)
#define EPS_F32  1.1920929e-07f

// ---------------- GEMM tiling ----------------
#define BM 128                    // rows per block (8 waves x 16 rows)
#define KB 32                     // K chunk staged in LDS
#define XS_PAD 36                 // x tile row pitch in floats (16B aligned stores)
#define WP_PITCH 66               // wg pair-tile row pitch in float2 units

typedef float v2f  __attribute__((ext_vector_type(2)));
typedef float v8f  __attribute__((ext_vector_type(8)));
typedef float v4f  __attribute__((ext_vector_type(4)));
// reduced-alignment vector for the (out+1)-offset output regions
typedef float v4fu __attribute__((ext_vector_type(4), aligned(4)));

// =====================================================================
// K1: logits[8192,64] = x[8192,1024] @ wg[1024,64] using fp32 WMMA.
//  - x tile staged via GLOBAL_LOAD_ASYNC_TO_LDS_B128 (ASYNCcnt path),
//    overlapped with synchronous wg-tile staging stores.
//  - wg tile stored as interleaved K-pairs so each B fragment is one
//    contiguous ds_load_b64 into an even VGPR pair (no v_mov fixups).
// =====================================================================
__global__ __launch_bounds__(256) void moe_gemm_wmma(const float* __restrict__ x,
                                                     const float* __restrict__ wg,
                                                     float* __restrict__ logits) {
    __shared__ float xs[BM * XS_PAD];              // 18432 B
    __shared__ v2f   wgp[(KB / 2) * WP_PITCH];     //  8448 B, {k even, k odd} pairs
    const int t    = threadIdx.x;
    const int wave = t >> 5;
    const int lane = t & 31;
    const int m    = lane & 15;                    // A row / B col within 16
    const int hi   = lane >> 4;                    // half-wave selects K pair
    const int row0 = blockIdx.x * BM;

    const unsigned xs_base = (unsigned)(uintptr_t)xs;   // low 32 bits = LDS offset

    v8f acc[4] = {};                               // 4 n-tiles of 16 cols

    for (int kb = 0; kb < D_MODEL; kb += KB) {
        // ---- async DMA: x tile [128 x 32] global -> LDS (no VGPR data) ----
        {
            const int r = t >> 3;              // 0..31
            const int c = (t & 7) << 2;        // 0..28 step 4
#pragma unroll
            for (int j = 0; j < 4; ++j) {
                const float* src = x + (size_t)(row0 + r + j * 32) * D_MODEL + kb + c;
                const unsigned dst = xs_base + (unsigned)(((r + j * 32) * XS_PAD + c) * 4);
                asm volatile("global_load_async_to_lds_b128 %0, %1, off"
                             :: "v"(dst), "v"((unsigned long long)(uintptr_t)src)
                             : "memory");
            }
        }
        // ---- wg tile [32 x 64] -> LDS as K-pairs (overlaps async DMA) ----
        {
#pragma unroll
            for (int j = 0; j < 2; ++j) {
                const int lin = t + j * 256;   // 0..511 float4s
                const int r   = lin >> 4;      // wg row (k) 0..31
                const int c   = (lin & 15) << 2;
                const float4 v = *(const float4*)(wg + (size_t)(kb + r) * N_EXPERT + c);
                float* dst = (float*)&wgp[(r >> 1) * WP_PITCH + c];
                const int sel = r & 1;
                dst[0 * 2 + sel] = v.x;
                dst[1 * 2 + sel] = v.y;
                dst[2 * 2 + sel] = v.z;
                dst[3 * 2 + sel] = v.w;
            }
        }
        if (kb + KB < D_MODEL) {  // pull the next x tile toward the caches
            __builtin_prefetch(x + (size_t)(row0 + wave * 16 + m) * D_MODEL + kb + KB, 0, 1);
        }
        asm volatile("s_wait_asynccnt 0x0" ::: "memory");
        __syncthreads();

        const int ra = wave * 16 + m;
#pragma unroll
        for (int kk = 0; kk < KB; kk += 4) {
            // A frag: K = kk+2*hi, kk+2*hi+1 (contiguous -> ds_load_b64)
            const v2f a = *(const v2f*)(xs + ra * XS_PAD + kk + 2 * hi);
            const int bp = ((kk >> 1) + hi) * WP_PITCH + m;
#pragma unroll
            for (int nt = 0; nt < 4; ++nt) {
                const v2f b = wgp[bp + nt * 16];   // contiguous pair -> ds_load_b64
#if __has_builtin(__builtin_amdgcn_wmma_f32_16x16x4_f32)
                acc[nt] = __builtin_amdgcn_wmma_f32_16x16x4_f32(
                    /*neg_a=*/false, a, /*neg_b=*/false, b,
                    /*c_mod=*/(short)0, acc[nt], /*reuse_a=*/false, /*reuse_b=*/false);
#else
#pragma unroll
                for (int r = 0; r < 8; ++r) acc[nt][r] += a.x * b.x + a.y * b.y;
#endif
            }
        }
        __syncthreads();
    }

    // C/D layout: VGPR r -> (M = r + 8*hi, N = lane&15) per n-tile
#pragma unroll
    for (int nt = 0; nt < 4; ++nt)
#pragma unroll
        for (int r = 0; r < 8; ++r)
            logits[(size_t)(row0 + wave * 16 + r + 8 * hi) * N_EXPERT + nt * 16 + m] = acc[nt][r];
}

// =====================================================================
// K2: per-row softmax + top-1/top-2 (one wave32 per row, 2 experts/lane)
// =====================================================================
__global__ __launch_bounds__(256) void moe_softmax_top2(const float* __restrict__ logits,
                                                        float* __restrict__ org_gates,
                                                        int* __restrict__ e1w, int* __restrict__ e2w,
                                                        float* __restrict__ g1w, float* __restrict__ g2w) {
    const int wave = threadIdx.x >> 5;
    const int lane = threadIdx.x & 31;
    const int row  = blockIdx.x * 8 + wave;

    const float z0 = logits[row * N_EXPERT + lane];
    const float z1 = logits[row * N_EXPERT + lane + 32];

    float mx = fmaxf(z0, z1);
    for (int off = 16; off; off >>= 1) mx = fmaxf(mx, __shfl_xor(mx, off, 32));
    const float ex0 = __expf(z0 - mx), ex1 = __expf(z1 - mx);
    float sm = ex0 + ex1;
    for (int off = 16; off; off >>= 1) sm += __shfl_xor(sm, off, 32);
    const float g0  = ex0 / sm;
    const float g32 = ex1 / sm;
    org_gates[row * N_EXPERT + lane]      = g0;
    org_gates[row * N_EXPERT + lane + 32] = g32;

    // top-1 (argmax on logits == argmax on gates; first-index tie-break)
    float bv = z0; int bi = lane;
    if (z1 > bv) { bv = z1; bi = lane + 32; }
    for (int off = 16; off; off >>= 1) {
        const float ov = __shfl_xor(bv, off, 32);
        const int   oi = __shfl_xor(bi, off, 32);
        if (ov > bv || (ov == bv && oi < bi)) { bv = ov; bi = oi; }
    }
    const int e1 = bi;

    // top-2: mask top-1 to -inf, repeat
    const float z0m = (e1 == lane)      ? -__builtin_inff() : z0;
    const float z1m = (e1 == lane + 32) ? -__builtin_inff() : z1;
    bv = z0m; bi = lane;
    if (z1m > bv) { bv = z1m; bi = lane + 32; }
    for (int off = 16; off; off >>= 1) {
        const float ov = __shfl_xor(bv, off, 32);
        const int   oi = __shfl_xor(bi, off, 32);
        if (ov > bv || (ov == bv && oi < bi)) { bv = ov; bi = oi; }
    }
    const int e2 = bi;

    const float g1 = __shfl((e1 >= 32) ? g32 : g0, e1 & 31, 32);
    const float g2 = __shfl((e2 >= 32) ? g32 : g0, e2 & 31, 32);
    if (lane == 0) { e1w[row] = e1; e2w[row] = e2; g1w[row] = g1; g2w[row] = g2; }
}

// =====================================================================
// K3a: per-chunk per-expert histograms (LDS atomics); 64 chunks x 128 rows
// =====================================================================
__global__ __launch_bounds__(128) void moe_hist(const int* __restrict__ e1w, const int* __restrict__ e2w,
                                                int* __restrict__ hist1, int* __restrict__ hist2) {
    __shared__ int h1[N_EXPERT], h2[N_EXPERT];
    const int t = threadIdx.x, b = blockIdx.x;
    if (t < N_EXPERT) { h1[t] = 0; h2[t] = 0; }
    __syncthreads();
    const int s = b * 128 + t;
    atomicAdd(&h1[e1w[s]], 1);
    atomicAdd(&h2[e2w[s]], 1);
    __syncthreads();
    if (t < N_EXPERT) { hist1[t * 64 + b] = h1[t]; hist2[t * 64 + b] = h2[t]; }
}

// =====================================================================
// K3b: per-expert scan over chunks + exp_counts + l_aux
// =====================================================================
__global__ __launch_bounds__(64) void moe_scan_stats(const int* __restrict__ hist1, const int* __restrict__ hist2,
                                                     int* __restrict__ base1, int* __restrict__ base2,
                                                     const float* __restrict__ org_gates,
                                                     float* __restrict__ out_laux,
                                                     float* __restrict__ out_expcounts) {
    __shared__ float red[N_EXPERT];
    const int e = threadIdx.x;
    int run = 0;
    for (int b = 0; b < 64; ++b) { base1[e * 64 + b] = run; run += hist1[e * 64 + b]; }
    const int total1 = run;                               // sum(mask1[:,e])
    for (int b = 0; b < 64; ++b) { base2[e * 64 + b] = run; run += hist2[e * 64 + b]; }
    out_expcounts[e] = (float)run;                        // total1 + total2

    float acc = 0.f;                                      // me = mean(gates, axis=0)
    for (int s = 0; s < S_TOKENS; ++s) acc += org_gates[s * N_EXPERT + e];
    const float me = acc / (float)S_TOKENS;
    const float ce = (float)total1 / (float)S_TOKENS;
    red[e] = me * ce * (float)(N_EXPERT * N_EXPERT);
    __syncthreads();
    if (e == 0) {
        float l = 0.f;
        for (int i = 0; i < N_EXPERT; ++i) l += red[i];
        *out_laux = l;
    }
}

// =====================================================================
// K3c: in-order rank assignment within each chunk; capacity drop -> -1
// =====================================================================
__global__ __launch_bounds__(64) void moe_rank(const int* __restrict__ e1w, const int* __restrict__ e2w,
                                               const int* __restrict__ base1, const int* __restrict__ base2,
                                               int* __restrict__ loc1, int* __restrict__ loc2) {
    const int e = threadIdx.x, b = blockIdx.x;
    int c1 = base1[e * 64 + b];
    int c2 = base2[e * 64 + b];
    for (int i = 0; i < 128; ++i) {
        const int s = b * 128 + i;           // broadcast reads across the 64 threads
        if (e1w[s] == e) { loc1[s] = (c1 < CAPACITY) ? c1 : -1; ++c1; }
        if (e2w[s] == e) { loc2[s] = (c2 < CAPACITY) ? c2 : -1; ++c2; }
    }
}

// =====================================================================
// K4: materialize combine_weights and dispatch_mask [S,E,C] — the
//     bandwidth roofline of this op (~1 GB of write-once data).
//     Non-temporal stores keep it from sweeping the 192MB L2; values are
//     computed during the zeroing pass (race-free single pass); aligned(4)
//     vectors because the region starts at out+1 (4B-aligned only).
// =====================================================================
__global__ __launch_bounds__(256) void moe_combine(const int* __restrict__ e1w, const int* __restrict__ e2w,
                                                   const int* __restrict__ loc1w, const int* __restrict__ loc2w,
                                                   const float* __restrict__ g1w, const float* __restrict__ g2w,
                                                   float* __restrict__ combine, float* __restrict__ dispatch) {
    const int s = blockIdx.x;
    const int t = threadIdx.x;
    const int l1 = loc1w[s], l2 = loc2w[s];
    const float g1 = (l1 >= 0) ? g1w[s] : 0.f;      // post-capacity-drop gates
    const float g2 = (l2 >= 0) ? g2w[s] : 0.f;
    const float denom = fmaxf(g1 + g2, EPS_F32);
    const float w1 = g1 / denom, w2 = g2 / denom;
    const int slot1 = (l1 >= 0) ? (e1w[s] * CAPACITY + l1) : -1;
    const int slot2 = (l2 >= 0) ? (e2w[s] * CAPACITY + l2) : -1;

    float* crow = combine  + (size_t)s * (N_EXPERT * CAPACITY);
    float* drow = dispatch + (size_t)s * (N_EXPERT * CAPACITY);
#pragma unroll
    for (int j = 0; j < 16; ++j) {
        const int q = j * 1024 + t * 4;      // coalesced 16B per thread
        v4fu cv = {}, dv = {};
#pragma unroll
        for (int k = 0; k < 4; ++k) {
            float v = 0.f;
            if (q + k == slot1)      v = w1;
            else if (q + k == slot2) v = w2;
            cv[k] = v;
            dv[k] = (v != 0.f) ? 1.f : 0.f;
        }
        __builtin_nontemporal_store(cv, (v4fu*)(crow + q));
        __builtin_nontemporal_store(dv, (v4fu*)(drow + q));
    }
}

// =====================================================================
extern "C" void kernel_launch(void* const* d_in, const int* in_sizes, int n_in,
                              void* d_out, int out_size, void* d_ws, size_t ws_size,
                              hipStream_t stream) {
    (void)in_sizes; (void)n_in; (void)out_size; (void)ws_size;
    const float* x  = (const float*)d_in[0];   // [8192,1024]
    const float* wg = (const float*)d_in[1];   // [1024,64]
    float* out = (float*)d_out;

    // output layout (return order): l_aux, combine, dispatch, exp_counts, org_gates
    const size_t SEC = (size_t)S_TOKENS * N_EXPERT * CAPACITY;
    float* out_laux      = out;
    float* out_combine   = out + 1;
    float* out_dispatch  = out_combine + SEC;
    float* out_expcounts = out_dispatch + SEC;
    float* out_orggates  = out_expcounts + N_EXPERT;

    // workspace layout (~2.4 MB)
    char* w = (char*)d_ws;
    float* logits = (float*)w;  w += sizeof(float) * (size_t)S_TOKENS * N_EXPERT;
    int*   e1w    = (int*)w;    w += sizeof(int) * S_TOKENS;
    int*   e2w    = (int*)w;    w += sizeof(int) * S_TOKENS;
    float* g1w    = (float*)w;  w += sizeof(float) * S_TOKENS;
    float* g2w    = (float*)w;  w += sizeof(float) * S_TOKENS;
    int*   loc1w  = (int*)w;    w += sizeof(int) * S_TOKENS;
    int*   loc2w  = (int*)w;    w += sizeof(int) * S_TOKENS;
    int*   hist1  = (int*)w;    w += sizeof(int) * N_EXPERT * 64;
    int*   hist2  = (int*)w;    w += sizeof(int) * N_EXPERT * 64;
    int*   base1  = (int*)w;    w += sizeof(int) * N_EXPERT * 64;
    int*   base2  = (int*)w;    w += sizeof(int) * N_EXPERT * 64;

    moe_gemm_wmma   <<<S_TOKENS / BM, 256, 0, stream>>>(x, wg, logits);
    moe_softmax_top2<<<S_TOKENS / 8, 256, 0, stream>>>(logits, out_orggates, e1w, e2w, g1w, g2w);
    moe_hist        <<<64, 128, 0, stream>>>(e1w, e2w, hist1, hist2);
    moe_scan_stats  <<<1, 64, 0, stream>>>(hist1, hist2, base1, base2, out_orggates, out_laux, out_expcounts);
    moe_rank        <<<64, 64, 0, stream>>>(e1w, e2w, base1, base2, loc1w, loc2w);
    moe_combine     <<<S_TOKENS, 256, 0, stream>>>(e1w, e2w, loc1w, loc2w, g1w, g2w, out_combine, out_dispatch);
}